// VoxelGridMaterial_29884382445935
// MI455X (gfx1250) — compile-verified
//
#include <hip/hip_runtime.h>
#include <stdint.h>

typedef float v4f __attribute__((ext_vector_type(4)));

#define BLOCK 256
#define TILES_PER_BLOCK 4
#define VDIM 128

// Trilinear voxel-grid interpolation, tuned for MI455X:
//  - 64MB grid stays L2-resident (RT loads), ipos/out streamed NT
//  - gfx1250 async global->LDS double-buffered prefetch of query positions
//  - b128 gathers, float4 FMA accumulate (VOPD-friendly), wave32 blocks
__global__ __launch_bounds__(BLOCK)
void voxel_trilerp(const float* __restrict__ ipos,
                   const float* __restrict__ grid,
                   float* __restrict__ out)
{
    __shared__ float sbuf[2][3 * BLOCK];   // 6 KB: double buffer of 256 float3

    const int tid   = threadIdx.x;
    const int tile0 = blockIdx.x * TILES_PER_BLOCK;
    const uint64_t ipos_base = (uint64_t)(uintptr_t)ipos;
    const v4f* __restrict__ g4 = (const v4f*)grid;   // 2 x v4f per voxel record

    // Async-copy one tile (256 queries = 768 floats) of ipos into LDS buffer b.
    // offset: is added to BOTH the LDS and global address, so one vdst/vaddr
    // pair covers all three 1KB-strided dwords per lane.
    auto issue_tile = [&](int tile, int b) {
        uint32_t ldsa = (uint32_t)(uintptr_t)(&sbuf[b][0]) + (uint32_t)tid * 4u;
        uint32_t voff = (uint32_t)(tile * (3 * BLOCK) + tid) * 4u;
        asm volatile(
            "global_load_async_to_lds_b32 %0, %1, %2 offset:0 th:TH_LOAD_NT\n\t"
            "global_load_async_to_lds_b32 %0, %1, %2 offset:1024 th:TH_LOAD_NT\n\t"
            "global_load_async_to_lds_b32 %0, %1, %2 offset:2048 th:TH_LOAD_NT"
            :: "v"(ldsa), "v"(voff), "s"(ipos_base)
            : "memory");
    };

    issue_tile(tile0, 0);

    for (int i = 0; i < TILES_PER_BLOCK; ++i) {
        const int buf = i & 1;
        if (i + 1 < TILES_PER_BLOCK) {
            issue_tile(tile0 + i + 1, buf ^ 1);
            // async loads retire in order: <=3 outstanding means the current
            // buffer's 3 loads are done while next tile's 3 stay in flight
            asm volatile("s_wait_asynccnt 3" ::: "memory");
        } else {
            asm volatile("s_wait_asynccnt 0" ::: "memory");
        }
        __syncthreads();

        float px = sbuf[buf][3 * tid + 0];
        float py = sbuf[buf][3 * tid + 1];
        float pz = sbuf[buf][3 * tid + 2];

        px = fminf(fmaxf(px, 0.0f), 1.0f) * (float)(VDIM - 1);
        py = fminf(fmaxf(py, 0.0f), 1.0f) * (float)(VDIM - 1);
        pz = fminf(fmaxf(pz, 0.0f), 1.0f) * (float)(VDIM - 1);

        int x0 = (int)px; x0 = x0 > (VDIM - 2) ? (VDIM - 2) : x0;  // p>=0: trunc==floor
        int y0 = (int)py; y0 = y0 > (VDIM - 2) ? (VDIM - 2) : y0;
        int z0 = (int)pz; z0 = z0 > (VDIM - 2) ? (VDIM - 2) : z0;

        const float fx = px - (float)x0, fy = py - (float)y0, fz = pz - (float)z0;
        const float gx = 1.0f - fx, gy = 1.0f - fy, gz = 1.0f - fz;

        const float wyz00 = gy * gz, wyz01 = gy * fz;
        const float wyz10 = fy * gz, wyz11 = fy * fz;

        // voxel linear index; steps: z=+1, y=+128, x=+16384 (voxel units)
        const int vbase = (x0 << 14) + (y0 << 7) + z0;

        v4f a0 = {0.f, 0.f, 0.f, 0.f};
        v4f a1 = {0.f, 0.f, 0.f, 0.f};

        auto corner = [&](int off, float w) {
            const v4f* c = g4 + 2 * (size_t)(vbase + off);
            v4f c0 = c[0], c1 = c[1];           // 8 params = 2 x b128 loads
            a0 += w * c0;
            a1 += w * c1;
        };
        // same corner order as reference (z fastest)
        corner(0,     gx * wyz00);
        corner(1,     gx * wyz01);
        corner(128,   gx * wyz10);
        corner(129,   gx * wyz11);
        corner(16384, fx * wyz00);
        corner(16385, fx * wyz01);
        corner(16512, fx * wyz10);
        corner(16513, fx * wyz11);

        const size_t q = (size_t)(tile0 + i) * BLOCK + tid;
        v4f* o = (v4f*)(out + q * 8);
        __builtin_nontemporal_store(a0, o);      // stream out, keep grid in L2
        __builtin_nontemporal_store(a1, o + 1);

        __syncthreads();   // protect LDS buffer before it is overwritten
    }
}

extern "C" void kernel_launch(void* const* d_in, const int* in_sizes, int n_in,
                              void* d_out, int out_size, void* d_ws, size_t ws_size,
                              hipStream_t stream) {
    (void)n_in; (void)out_size; (void)d_ws; (void)ws_size;
    const float* ipos = (const float*)d_in[0];   // (NQ, 3) f32
    const float* grid = (const float*)d_in[1];   // (128,128,128,8) f32
    float* out = (float*)d_out;                  // (NQ, 8) f32

    const int nq     = in_sizes[0] / 3;                    // 4,194,304
    const int blocks = nq / (BLOCK * TILES_PER_BLOCK);     // 4096 (exact fit)
    voxel_trilerp<<<blocks, BLOCK, 0, stream>>>(ipos, grid, out);
}